// SimplifiedAWBM_41317585387786
// MI455X (gfx1250) — compile-verified
//
#include <hip/hip_runtime.h>
#include <stdint.h>

// ---------------------------------------------------------------------------
// AWBM sequential scan, parallelized as a 3-level function-composition scan.
//   S_t = min(max(S_{t-1} + p_t - e_t, 0), Smax)      (clamp-add monoid, 3 floats)
//   excess_t = max(S'_t - Smax, 0)
//   B_t = k*B_{t-1} + k*bfi*excess_t                  (affine monoid, 2 floats)
//   total_t = (1-bfi)*excess_t + (1-k)*(B_{t-1} + bfi*excess_t)
//
// Pipeline: block S-aggregates -> parallel scan -> block B-aggregates ->
// parallel scan -> output pass. x (64MB) fits in the 192MB L2, so re-reads in
// later passes hit L2; HBM traffic stays ~96MB (~4.1us roofline at 23.3TB/s).
// WMMA intentionally unused: ~1 flop/byte, no dense matmul structure.
//
// CDNA5 specifics: TDM tensor_load_to_lds (TENSORcnt) double-buffered 64KB
// tiles; TDM pad feature (+8B per 256B row) so per-lane ds_load_b64 rows hit
// distinct bank pairs; tensor_dim0 OOB zero-fill for the ragged final tile;
// s_wait_tensorcnt + workgroup barriers for publication.
// ---------------------------------------------------------------------------

#define THREADS 256
#define SEG 32                      // steps per thread per tile
#define TILE (THREADS * SEG)        // 8192 steps per tile (64KB of x)
#define NTILES 4
#define CHUNK (TILE * NTILES)       // 32768 steps per block
#define ROW_DW 66                   // 64 data dwords + 2 pad dwords per thread-row
#define FLT_BIG 3.0e38f

typedef __attribute__((ext_vector_type(4))) unsigned int u32x4;
typedef __attribute__((ext_vector_type(8))) int i32x8;
typedef __attribute__((ext_vector_type(4))) int i32x4;

struct ClampFn { float a, b, c; };   // f(x) = min(max(x + a, b), c)

__device__ __forceinline__ ClampFn fn_compose(ClampFn older, ClampFn newer) {
  ClampFn r;
  r.a = older.a + newer.a;
  r.b = fminf(fmaxf(older.b + newer.a, newer.b), newer.c);
  r.c = fminf(fmaxf(older.c + newer.a, newer.b), newer.c);
  return r;
}
__device__ __forceinline__ float fn_eval(ClampFn f, float x) {
  return fminf(fmaxf(x + f.a, f.b), f.c);
}

// Issue one TDM tile load: `valid` fp32 elements are readable starting at gsrc;
// reads past that zero-fill. LDS gets 2 pad dwords after every 64 data dwords.
__device__ __forceinline__ void tdm_load_tile(const float* gsrc, unsigned valid,
                                              unsigned lds_byte_off) {
  uint64_t ga = (uint64_t)(uintptr_t)gsrc;
  u32x4 g0;
  g0[0] = 1u;                                        // count=1, user descriptor
  g0[1] = lds_byte_off;                              // lds_addr
  g0[2] = (unsigned)(ga & 0xFFFFFFFFu);              // global_addr[31:0]
  g0[3] = (unsigned)((ga >> 32) & 0x1FFFFFFu)        // global_addr[56:32]
        | (2u << 30);                                // type = 2 ("image")

  unsigned td0 = valid;                              // tensor_dim0 (OOB bound)
  unsigned stride0 = valid ? valid : 1u;
  i32x8 g1;
  g1[0] = (int)((2u << 16)      // data_size = 4B
        | (1u << 20)            // pad_enable
        | (5u << 22)            // pad_interval: 64 dwords (256B)
        | (1u << 25));          // pad_amount: 2 dwords (8B)
  g1[1] = (int)((td0 & 0xFFFFu) << 16);              // tensor_dim0[15:0]
  g1[2] = (int)(((td0 >> 16) & 0xFFFFu)              // tensor_dim0[31:16]
        | (1u << 16));                               // tensor_dim1 = 1
  g1[3] = (int)((unsigned)(TILE * 2) << 16);         // tile_dim0 = 16384 floats
  g1[4] = (int)1u;                                   // tile_dim1 = 1
  g1[5] = (int)stride0;                              // tensor_dim0_stride[31:0]
  g1[6] = 0;
  g1[7] = 0;
  i32x4 z4 = {0, 0, 0, 0};
#if defined(__clang_major__) && (__clang_major__ >= 23)
  i32x8 z8 = {0, 0, 0, 0, 0, 0, 0, 0};
  __builtin_amdgcn_tensor_load_to_lds(g0, g1, z4, z4, z8, 0);
#else
  __builtin_amdgcn_tensor_load_to_lds(g0, g1, z4, z4, 0);
#endif
}

// mode 0: write per-block S clamp-function aggregates  -> ws[0 .. 3N)
// mode 1: read S block-starts, write per-block affine (m,w) -> ws[4N .. 6N)
// mode 2: read S and B block-starts, write totalflow outputs
__global__ __launch_bounds__(THREADS) void awbm_block_kernel(
    const float* __restrict__ x, const float* __restrict__ pBFI,
    const float* __restrict__ pK, const float* __restrict__ pSmax,
    float* __restrict__ out, float* __restrict__ ws,
    int T, int nblock, int mode) {
  extern __shared__ float smem[];
  float* ldsx = smem;                                 // 2 * 256*66 floats
  float* sa = smem + 2 * THREADS * ROW_DW;
  float* sb = sa + THREADS;
  float* sc = sb + THREADS;
  float* sm = sc + THREADS;
  float* sw = sm + THREADS;

  const int tid = threadIdx.x;
  const int bid = blockIdx.x;
  const float bfi = pBFI[0], k = pK[0], smax = pSmax[0];
  const float kbfi = k * bfi;
  float k32 = k * k; k32 *= k32; k32 *= k32; k32 *= k32; k32 *= k32;  // k^32

  const int blockStart = bid * CHUNK;
  float S_run = 0.0f, B_run = 0.0f;
  if (mode >= 1) S_run = ws[3 * nblock + bid];
  if (mode == 2) B_run = ws[6 * nblock + bid];

  ClampFn blockFn = {0.0f, -FLT_BIG, FLT_BIG};
  float blkM = 1.0f, blkW = 0.0f;

  const unsigned ldsBase = (unsigned)(uintptr_t)&ldsx[0];
  const unsigned bufBytes = (unsigned)(THREADS * ROW_DW * 4);

  // Prologue: DMA tile 0, wait, publish.
  if (tid == 0) {
    int rem = T - blockStart; if (rem < 0) rem = 0;
    tdm_load_tile(x + (size_t)blockStart * 2, (unsigned)rem * 2u, ldsBase);
    __builtin_amdgcn_s_wait_tensorcnt(0);
  }
  __syncthreads();

  for (int tile = 0; tile < NTILES; ++tile) {
    // Overlap: DMA next tile into the other buffer while computing this one.
    if (tid == 0 && tile + 1 < NTILES) {
      int s1 = blockStart + (tile + 1) * TILE;
      int rem = T - s1; if (rem < 0) rem = 0;
      tdm_load_tile(x + (size_t)s1 * 2, (unsigned)rem * 2u,
                    ldsBase + (unsigned)((tile + 1) & 1) * bufBytes);
    }

    const float* xb = ldsx + (size_t)(tile & 1) * (THREADS * ROW_DW);
    const float2* row = (const float2*)(xb + tid * ROW_DW);  // padded rows

    // Phase 1: read deltas once into registers; build the clamp-function.
    float d[SEG];
    ClampFn f = {0.0f, -FLT_BIG, FLT_BIG};
#pragma unroll
    for (int j = 0; j < SEG; ++j) {
      float2 pe = row[j];
      d[j] = pe.x - pe.y;
      f.a += d[j];
      f.b = fminf(fmaxf(f.b + d[j], 0.0f), smax);
      f.c = fminf(fmaxf(f.c + d[j], 0.0f), smax);
    }

    // Inclusive Hillis-Steele composition scan across 256 threads.
    sa[tid] = f.a; sb[tid] = f.b; sc[tid] = f.c;
    __syncthreads();
    ClampFn cur = f;
    for (int off = 1; off < THREADS; off <<= 1) {
      ClampFn p; bool has = (tid >= off);
      if (has) { p.a = sa[tid - off]; p.b = sb[tid - off]; p.c = sc[tid - off]; }
      __syncthreads();
      if (has) {
        cur = fn_compose(p, cur);
        sa[tid] = cur.a; sb[tid] = cur.b; sc[tid] = cur.c;
      }
      __syncthreads();
    }
    ClampFn tileFn = {sa[THREADS - 1], sb[THREADS - 1], sc[THREADS - 1]};

    if (mode == 0) {
      blockFn = fn_compose(blockFn, tileFn);
    } else {
      // S at this thread's segment start (exclusive prefix applied to S_run).
      float S;
      if (tid == 0) {
        S = S_run;
      } else {
        ClampFn pf = {sa[tid - 1], sb[tid - 1], sc[tid - 1]};
        S = fn_eval(pf, S_run);
      }

      // Phase 2: S sequentially from registers -> excess + affine aggregate.
      float w = 0.0f;
      float ex[SEG];
#pragma unroll
      for (int j = 0; j < SEG; ++j) {
        S = fmaxf(S + d[j], 0.0f);
        float excess = fmaxf(S - smax, 0.0f);
        S -= excess;
        ex[j] = excess;
        w = k * w + kbfi * excess;              // B' = k*B + k*bfi*excess
      }

      sm[tid] = k32; sw[tid] = w;
      __syncthreads();
      float cm = k32, cw = w;
      for (int off = 1; off < THREADS; off <<= 1) {
        float pm = 1.0f, pw = 0.0f; bool has = (tid >= off);
        if (has) { pm = sm[tid - off]; pw = sw[tid - off]; }
        __syncthreads();
        if (has) {
          cw = pw * cm + cw;
          cm = pm * cm;
          sm[tid] = cm; sw[tid] = cw;
        }
        __syncthreads();
      }
      float tileM = sm[THREADS - 1], tileW = sw[THREADS - 1];

      if (mode == 1) {
        blkW = blkW * tileM + tileW;            // block affine = tile ∘ block
        blkM = blkM * tileM;
      } else {
        float B = (tid == 0) ? B_run : (sm[tid - 1] * B_run + sw[tid - 1]);
        int gstep = blockStart + tile * TILE + tid * SEG;
#pragma unroll
        for (int j = 0; j < SEG; ++j) {
          float excess = ex[j];
          B = B + bfi * excess;
          float baseflow = (1.0f - k) * B;
          B = B - baseflow;                     // B = k*(B_prev + bfi*excess)
          float total = (1.0f - bfi) * excess + baseflow;
          int t = gstep + j;
          if (t < T) out[t] = total;
        }
        B_run = tileM * B_run + tileW;
      }
      S_run = fn_eval(tileFn, S_run);
    }

    // Wait for next tile's DMA, then republish LDS to all waves.
    if (tid == 0 && tile + 1 < NTILES) __builtin_amdgcn_s_wait_tensorcnt(0);
    __syncthreads();
  }

  if (tid == 0) {
    if (mode == 0) {
      ws[3 * bid + 0] = blockFn.a;
      ws[3 * bid + 1] = blockFn.b;
      ws[3 * bid + 2] = blockFn.c;
    } else if (mode == 1) {
      ws[4 * nblock + 2 * bid + 0] = blkM;
      ws[4 * nblock + 2 * bid + 1] = blkW;
    }
  }
}

// Parallel inter-block scans (critical-path fix: was 245 serially-dependent
// global loads ~100us; now coalesced load + 8 LDS log-rounds).
__global__ __launch_bounds__(THREADS) void awbm_scan_s_kernel(
    float* __restrict__ ws, int nblock) {
  __shared__ float ga[THREADS], gb[THREADS], gc[THREADS];
  const int tid = threadIdx.x;
  ClampFn carry = {0.0f, -FLT_BIG, FLT_BIG};          // identity
  for (int base = 0; base < nblock; base += THREADS) {
    const int i = base + tid;
    ClampFn f = {0.0f, -FLT_BIG, FLT_BIG};
    if (i < nblock) { f.a = ws[3 * i]; f.b = ws[3 * i + 1]; f.c = ws[3 * i + 2]; }
    ga[tid] = f.a; gb[tid] = f.b; gc[tid] = f.c;
    __syncthreads();
    ClampFn cur = f;
    for (int off = 1; off < THREADS; off <<= 1) {
      ClampFn p; bool has = (tid >= off);
      if (has) { p.a = ga[tid - off]; p.b = gb[tid - off]; p.c = gc[tid - off]; }
      __syncthreads();
      if (has) {
        cur = fn_compose(p, cur);
        ga[tid] = cur.a; gb[tid] = cur.b; gc[tid] = cur.c;
      }
      __syncthreads();
    }
    ClampFn excl = carry;
    if (tid > 0) {
      ClampFn ip = {ga[tid - 1], gb[tid - 1], gc[tid - 1]};
      excl = fn_compose(carry, ip);
    }
    if (i < nblock) ws[3 * nblock + i] = fn_eval(excl, 0.5f);   // S_INIT
    ClampFn agg = {ga[THREADS - 1], gb[THREADS - 1], gc[THREADS - 1]};
    carry = fn_compose(carry, agg);
    __syncthreads();
  }
}

__global__ __launch_bounds__(THREADS) void awbm_scan_b_kernel(
    float* __restrict__ ws, int nblock) {
  __shared__ float gm[THREADS], gw[THREADS];
  const int tid = threadIdx.x;
  float carryM = 1.0f, carryW = 0.0f;                 // identity
  for (int base = 0; base < nblock; base += THREADS) {
    const int i = base + tid;
    float m = 1.0f, w = 0.0f;
    if (i < nblock) { m = ws[4 * nblock + 2 * i]; w = ws[4 * nblock + 2 * i + 1]; }
    gm[tid] = m; gw[tid] = w;
    __syncthreads();
    float cm = m, cw = w;
    for (int off = 1; off < THREADS; off <<= 1) {
      float pm = 1.0f, pw = 0.0f; bool has = (tid >= off);
      if (has) { pm = gm[tid - off]; pw = gw[tid - off]; }
      __syncthreads();
      if (has) {
        cw = pw * cm + cw;
        cm = pm * cm;
        gm[tid] = cm; gw[tid] = cw;
      }
      __syncthreads();
    }
    float exM = carryM, exW = carryW;
    if (tid > 0) {
      float im = gm[tid - 1], iw = gw[tid - 1];
      exW = carryW * im + iw;                         // carry then chunk prefix
      exM = carryM * im;
    }
    if (i < nblock) ws[6 * nblock + i] = exM * 1.0f + exW;      // B_INIT = 1
    float am = gm[THREADS - 1], aw = gw[THREADS - 1];
    carryW = carryW * am + aw;
    carryM = carryM * am;
    __syncthreads();
  }
}

extern "C" void kernel_launch(void* const* d_in, const int* in_sizes, int n_in,
                              void* d_out, int out_size, void* d_ws, size_t ws_size,
                              hipStream_t stream) {
  const float* x    = (const float*)d_in[0];
  const float* BFI  = (const float*)d_in[1];
  const float* K    = (const float*)d_in[2];
  const float* Smax = (const float*)d_in[3];
  float* out = (float*)d_out;
  float* ws  = (float*)d_ws;

  const int T = in_sizes[0] / 2;
  const int nblock = (T + CHUNK - 1) / CHUNK;              // 245 for T=8M
  const size_t smemBytes = (size_t)(2 * THREADS * ROW_DW + 5 * THREADS) * sizeof(float);

  dim3 grid(nblock), blk(THREADS);
  awbm_block_kernel<<<grid, blk, smemBytes, stream>>>(x, BFI, K, Smax, out, ws, T, nblock, 0);
  awbm_scan_s_kernel<<<1, blk, 0, stream>>>(ws, nblock);
  awbm_block_kernel<<<grid, blk, smemBytes, stream>>>(x, BFI, K, Smax, out, ws, T, nblock, 1);
  awbm_scan_b_kernel<<<1, blk, 0, stream>>>(ws, nblock);
  awbm_block_kernel<<<grid, blk, smemBytes, stream>>>(x, BFI, K, Smax, out, ws, T, nblock, 2);
}